// RetrievalTool_26938034881191
// MI455X (gfx1250) — compile-verified
//
#include <hip/hip_runtime.h>
#include <math.h>
#include <stdint.h>

#define CAT_DIM 16
#define CTX_DIM 64
#define GATE_HID 128
#define N_PERIOD 3
#define TOPM 20
#define B_SZ 64
#define N_POOL 200000
#define PRED_LEN 96
#define CHANNELS 8
#define LN_EPS 1e-5f
#define TEMP_INV 10.0f
#define ALPHA 0.7f

#define CHUNK 1024
#define SUBT 64
#define NSUB (CHUNK / SUBT)
#define NCHUNK ((N_POOL + CHUNK - 1) / CHUNK) /* 196 */
#define NCAND (NCHUNK * TOPM)                 /* 3920 */

#define QSTR 68
#define PSTR 68
#define SSTR 65

typedef float v2f __attribute__((ext_vector_type(2)));
typedef float v8f __attribute__((ext_vector_type(8)));

/* ---- CDNA5 async HBM->LDS path (GLOBAL_LOAD_ASYNC_TO_LDS_B128, ASYNCcnt) ----
 * The clang builtin's pointer params use the HIP cuda_device LangAS which is
 * not source-constructible, so issue the instruction via inline asm (ISA
 * §15.18: "global_load_async_to_lds_b128 vdst, v[addr], off", GV mode;
 * VDST holds the LDS byte address = low 32 bits of a generic LDS pointer). */
#define USE_ASYNC_LDS 1

#if USE_ASYNC_LDS
__device__ __forceinline__ void async_g2l_b128(const float* g, float* l) {
  asm volatile("global_load_async_to_lds_b128 %0, %1, off"
               :
               : "v"((uint32_t)(uintptr_t)l), "v"(g)
               : "memory");
}
__device__ __forceinline__ void wait_async0() {
  asm volatile("s_wait_asynccnt 0" ::: "memory");
}
#endif

__device__ __forceinline__ float gelu_exact(float x) {
  return 0.5f * x * (1.0f + erff(x * 0.70710678118654752f));
}

__device__ __forceinline__ float reduce64(float v, float* red, int t) {
  __syncthreads();
  if (t < 64) red[t] = v;
  __syncthreads();
  if (t < 32) red[t] += red[t + 32];
  __syncthreads();
  if (t < 16) red[t] += red[t + 16];
  __syncthreads();
  if (t < 8) red[t] += red[t + 8];
  __syncthreads();
  if (t < 4) red[t] += red[t + 4];
  __syncthreads();
  if (t < 2) red[t] += red[t + 2];
  __syncthreads();
  if (t < 1) red[t] += red[t + 1];
  __syncthreads();
  float r = red[0];
  __syncthreads();
  return r;
}

/* ---------------- Kernel 1: context encode -> q, qn (f32) ---------------- */
__global__ __launch_bounds__(128)
void encode_kernel(const int* __restrict__ dataset, const int* __restrict__ sensor,
                   const int* __restrict__ location, const int* __restrict__ hour,
                   const int* __restrict__ weekday, const int* __restrict__ month,
                   const int* __restrict__ holiday, const int* __restrict__ peak,
                   const float* __restrict__ local_state,
                   const float* __restrict__ emb_dataset, const float* __restrict__ emb_sensor,
                   const float* __restrict__ emb_location, const float* __restrict__ emb_hour,
                   const float* __restrict__ emb_weekday, const float* __restrict__ emb_month,
                   const float* __restrict__ emb_holiday, const float* __restrict__ emb_peak,
                   const float* __restrict__ cat_w1, const float* __restrict__ cat_b1,
                   const float* __restrict__ cat_w2, const float* __restrict__ cat_b2,
                   const float* __restrict__ loc_w1, const float* __restrict__ loc_b1,
                   const float* __restrict__ loc_w2, const float* __restrict__ loc_b2,
                   const float* __restrict__ ln_g, const float* __restrict__ ln_b,
                   float* __restrict__ q_ws, float* __restrict__ qn_ws)
{
  __shared__ float cat_s[128];
  __shared__ float h1_s[64];
  __shared__ float cat_o[64];
  __shared__ float lh_s[64];
  __shared__ float red[64];
  const int t = threadIdx.x;
  const int b = blockIdx.x;

  { /* categorical embedding concat: 8 x 16 = 128 */
    int e = t >> 4, c = t & 15;
    const float* emb; int id;
    switch (e) {
      case 0: emb = emb_dataset;  id = dataset[b]; break;
      case 1: emb = emb_sensor;   id = sensor[b]; break;
      case 2: emb = emb_location; id = location[b]; break;
      case 3: emb = emb_hour;     id = min(max(hour[b], 0), 23); break;
      case 4: emb = emb_weekday;  id = min(max(weekday[b], 0), 6); break;
      case 5: emb = emb_month;    id = min(max(month[b], 1), 12); break;
      case 6: emb = emb_holiday;  id = min(max(holiday[b], 0), 1); break;
      default: emb = emb_peak;    id = min(max(peak[b], 0), 1); break;
    }
    cat_s[t] = emb[id * CAT_DIM + c];
  }
  __syncthreads();
  if (t < 64) {
    float acc = cat_b1[t];
    for (int i = 0; i < 128; ++i) acc += cat_s[i] * cat_w1[i * 64 + t];
    h1_s[t] = gelu_exact(acc);
  }
  __syncthreads();
  if (t < 64) {
    float acc = cat_b2[t];
    for (int i = 0; i < 64; ++i) acc += h1_s[i] * cat_w2[i * 64 + t];
    cat_o[t] = acc;
  }
  __syncthreads();

  for (int p = 0; p < N_PERIOD; ++p) {
    float x0 = local_state[(b * N_PERIOD + p) * 4 + 0];
    float x1 = local_state[(b * N_PERIOD + p) * 4 + 1];
    float x2 = local_state[(b * N_PERIOD + p) * 4 + 2];
    float x3 = local_state[(b * N_PERIOD + p) * 4 + 3];
    if (t < 64) {
      float acc = loc_b1[t];
      acc += x0 * loc_w1[0 * 64 + t] + x1 * loc_w1[1 * 64 + t]
           + x2 * loc_w1[2 * 64 + t] + x3 * loc_w1[3 * 64 + t];
      lh_s[t] = gelu_exact(acc);
    }
    __syncthreads();
    float v = 0.0f;
    if (t < 64) {
      float acc = loc_b2[t];
      for (int i = 0; i < 64; ++i) acc += lh_s[i] * loc_w2[i * 64 + t];
      v = cat_o[t] + acc;
    }
    float mu = reduce64(t < 64 ? v : 0.0f, red, t) * (1.0f / 64.0f);
    float d = v - mu;
    float var = reduce64(t < 64 ? d * d : 0.0f, red, t) * (1.0f / 64.0f);
    float y = 0.0f;
    if (t < 64) y = d * rsqrtf(var + LN_EPS) * ln_g[t] + ln_b[t];
    float nsq = reduce64(t < 64 ? y * y : 0.0f, red, t);
    float inv = 1.0f / (sqrtf(nsq) + 1e-8f);
    if (t < 64) {
      size_t o = ((size_t)(p * B_SZ + b)) * CTX_DIM + t;
      q_ws[o] = y;
      qn_ws[o] = y * inv;
    }
    __syncthreads();
  }
}

/* ------- Kernel 2: cosine sim via FP32 WMMA + streaming per-chunk top-20 ------- */
__global__ __launch_bounds__(256)
void sim_topk_kernel(const float* __restrict__ pool_ctx,
                     const float* __restrict__ qn_ws,
                     float* __restrict__ sa_val, int* __restrict__ sa_idx)
{
  __shared__ float qn_s[64 * QSTR];
  __shared__ float pool_s[SUBT * PSTR];
  __shared__ float sim_s[64 * SSTR];
  __shared__ float rowsq[SUBT];
  __shared__ float tlv[64 * TOPM];
  __shared__ int   tli[64 * TOPM];

  const int t = threadIdx.x;
  const int lane = t & 31;
  const int wave = t >> 5;
  const int chunk = blockIdx.x;
  const int p = blockIdx.y;
  const int chunk_base = chunk * CHUNK;
  const float* poolp = pool_ctx + (size_t)p * N_POOL * CTX_DIM;

  /* stage qn tile [64 x 64] */
  for (int f = t; f < 64 * 16; f += 256) {
    int r = f >> 4, s = f & 15;
    float4 vv = *(const float4*)(qn_ws + ((size_t)(p * B_SZ + r)) * CTX_DIM + s * 4);
    *(float4*)(qn_s + r * QSTR + s * 4) = vv;
  }
  if (t < 64) {
    for (int m = 0; m < TOPM; ++m) { tlv[t * TOPM + m] = -3.0e38f; tli[t * TOPM + m] = 0; }
  }
  float kmin = -3.0e38f; int kslot = 0;
  __syncthreads();

  for (int sub = 0; sub < NSUB; ++sub) {
    const int base = chunk_base + sub * SUBT;
    if (t < SUBT) rowsq[t] = 0.0f;

#if USE_ASYNC_LDS
    /* direct HBM -> LDS, tracked by ASYNCcnt (no VGPR round-trip) */
    for (int f = t; f < SUBT * 16; f += 256) {
      int r = f >> 4, s = f & 15;
      int n = base + r;
      if (n < N_POOL)
        async_g2l_b128(poolp + (size_t)n * CTX_DIM + s * 4,
                       pool_s + r * PSTR + s * 4);
    }
    if (sub + 1 < NSUB) { /* prefetch next sub-tile toward L2 */
      int nn = base + SUBT + (t & 63);
      if (nn < N_POOL) __builtin_prefetch(poolp + (size_t)nn * CTX_DIM, 0, 1);
    }
    wait_async0();
    __syncthreads();
    /* per-row ||p||^2 from LDS read-back */
    for (int f = t; f < SUBT * 16; f += 256) {
      int r = f >> 4, s = f & 15;
      float4 vv = *(const float4*)(pool_s + r * PSTR + s * 4);
      atomicAdd(&rowsq[r], vv.x * vv.x + vv.y * vv.y + vv.z * vv.z + vv.w * vv.w);
    }
    __syncthreads();
#else
    __syncthreads();
    /* fallback: load 64 pool rows via VGPRs, accumulate per-row sum of squares */
    for (int f = t; f < SUBT * 16; f += 256) {
      int r = f >> 4, s = f & 15;
      int n = base + r;
      float4 vv = make_float4(0.f, 0.f, 0.f, 0.f);
      if (n < N_POOL) vv = *(const float4*)(poolp + (size_t)n * CTX_DIM + s * 4);
      *(float4*)(pool_s + r * PSTR + s * 4) = vv;
      atomicAdd(&rowsq[r], vv.x * vv.x + vv.y * vv.y + vv.z * vv.z + vv.w * vv.w);
    }
    if (sub + 1 < NSUB) {
      int nn = base + SUBT + (t & 63);
      if (nn < N_POOL) __builtin_prefetch(poolp + (size_t)nn * CTX_DIM, 0, 1);
    }
    __syncthreads();
#endif

    /* 4 b-tiles x 4 n-tiles of 16x16, K=64 via 16 chained f32 WMMAs */
    const int lm = lane & 15;
    const int koff = (lane < 16) ? 0 : 2;
    for (int i = 0; i < 2; ++i) {
      int tile = wave * 2 + i;
      int bt = tile >> 2, nt = tile & 3;
      const float* arow = qn_s + (bt * 16 + lm) * QSTR + koff;
      const float* brow = pool_s + (nt * 16 + lm) * PSTR + koff;
      v8f acc = {};
      #pragma unroll
      for (int k0 = 0; k0 < 64; k0 += 4) {
        v2f a;  a.x = arow[k0];  a.y = arow[k0 + 1];
        v2f bb; bb.x = brow[k0]; bb.y = brow[k0 + 1];
        acc = __builtin_amdgcn_wmma_f32_16x16x4_f32(false, a, false, bb,
                                                    (short)0, acc, false, false);
      }
      float invn = 1.0f / (sqrtf(rowsq[nt * 16 + lm]) + 1e-8f);
      int mh = (lane < 16) ? 0 : 8;
      #pragma unroll
      for (int r = 0; r < 8; ++r)
        sim_s[(bt * 16 + mh + r) * SSTR + nt * 16 + lm] = acc[r] * invn;
    }
    __syncthreads();

    /* streaming top-20: one thread per b row */
    if (t < 64) {
      const float* srow = sim_s + t * SSTR;
      int nvalid = min(SUBT, N_POOL - base);
      for (int j = 0; j < nvalid; ++j) {
        float v = srow[j];
        if (v > kmin) {
          tlv[t * TOPM + kslot] = v;
          tli[t * TOPM + kslot] = base + j;
          float mn = tlv[t * TOPM]; int ms = 0;
          #pragma unroll
          for (int s2 = 1; s2 < TOPM; ++s2) {
            float vv = tlv[t * TOPM + s2];
            if (vv < mn) { mn = vv; ms = s2; }
          }
          kmin = mn; kslot = ms;
        }
      }
    }
    __syncthreads();
  }

  if (t < 64) {
    size_t o = ((size_t)((p * NCHUNK + chunk) * B_SZ + t)) * TOPM;
    for (int m = 0; m < TOPM; ++m) {
      sa_val[o + m] = tlv[t * TOPM + m];
      sa_idx[o + m] = tli[t * TOPM + m];
    }
  }
}

/* ------- Kernel 3: merge chunk lists -> global top-20, softmax, gate MLP ------- */
__global__ __launch_bounds__(256)
void merge_gate_kernel(const float* __restrict__ sa_val, const int* __restrict__ sa_idx,
                       const float* __restrict__ q_ws,
                       const float* __restrict__ pool_ctx,
                       const float* __restrict__ gate_w1, const float* __restrict__ gate_b1,
                       const float* __restrict__ gate_w2, const float* __restrict__ gate_b2,
                       float* __restrict__ w_ws, int* __restrict__ ci_ws)
{
  __shared__ float sv[NCAND];
  __shared__ int   si[NCAND];
  __shared__ float rv[256];
  __shared__ int   rp[256];
  __shared__ float top_val[TOPM];
  __shared__ int   top_idx[TOPM];
  __shared__ float wsim[TOPM];
  __shared__ float gv[TOPM];
  __shared__ float qrow[64];
  __shared__ float crow[64];

  const int t = threadIdx.x;
  const int p = blockIdx.x / B_SZ;
  const int b = blockIdx.x % B_SZ;

  for (int e = t; e < NCAND; e += 256) {
    int c = e / TOPM, m = e % TOPM;
    size_t o = ((size_t)((p * NCHUNK + c) * B_SZ + b)) * TOPM + m;
    sv[e] = sa_val[o];
    si[e] = sa_idx[o];
  }
  if (t < 64) qrow[t] = q_ws[((size_t)(p * B_SZ + b)) * CTX_DIM + t];
  __syncthreads();

  for (int round = 0; round < TOPM; ++round) {
    float bv = -3.4e38f; int bp = 0;
    for (int e = t; e < NCAND; e += 256) {
      float v = sv[e];
      if (v > bv) { bv = v; bp = e; }
    }
    rv[t] = bv; rp[t] = bp;
    __syncthreads();
    for (int off = 128; off > 0; off >>= 1) {
      if (t < off && rv[t + off] > rv[t]) { rv[t] = rv[t + off]; rp[t] = rp[t + off]; }
      __syncthreads();
    }
    if (t == 0) {
      int pos = rp[0];
      top_val[round] = rv[0];
      top_idx[round] = si[pos];
      sv[pos] = -3.4e38f;
    }
    __syncthreads();
  }

  if (t == 0) {
    float mx = top_val[0]; /* sorted descending */
    float s = 0.0f;
    for (int m = 0; m < TOPM; ++m) { float e = expf((top_val[m] - mx) * TEMP_INV); wsim[m] = e; s += e; }
    float inv = 1.0f / s;
    for (int m = 0; m < TOPM; ++m) wsim[m] *= inv;
  }
  __syncthreads();

  const float* poolp = pool_ctx + (size_t)p * N_POOL * CTX_DIM;
  for (int m = 0; m < TOPM; ++m) {
    if (t < 64) crow[t] = poolp[(size_t)top_idx[m] * CTX_DIM + t];
    __syncthreads();
    float contrib = 0.0f;
    if (t < GATE_HID) {
      float acc = gate_b1[t];
      for (int i = 0; i < 64; ++i) acc += qrow[i] * gate_w1[i * GATE_HID + t];
      for (int i = 0; i < 64; ++i) acc += crow[i] * gate_w1[(64 + i) * GATE_HID + t];
      acc += top_val[m] * gate_w1[128 * GATE_HID + t];
      acc += wsim[m] * gate_w1[129 * GATE_HID + t];
      contrib = gelu_exact(acc) * gate_w2[t];
    }
    rv[t] = contrib;
    __syncthreads();
    for (int off = 128; off > 0; off >>= 1) {
      if (t < off) rv[t] += rv[t + off];
      __syncthreads();
    }
    if (t == 0) gv[m] = rv[0] + gate_b2[0];
    __syncthreads();
  }

  if (t == 0) {
    float mx = gv[0];
    for (int m = 1; m < TOPM; ++m) mx = fmaxf(mx, gv[m]);
    float s = 0.0f;
    for (int m = 0; m < TOPM; ++m) { float e = expf(gv[m] - mx); gv[m] = e; s += e; }
    float inv = 1.0f / s;
    for (int m = 0; m < TOPM; ++m) {
      float w = ALPHA * wsim[m] + (1.0f - ALPHA) * gv[m] * inv;
      size_t o = ((size_t)(p * B_SZ + b)) * TOPM + m;
      w_ws[o] = w;
      ci_ws[o] = top_idx[m];
    }
  }
}

/* ---------------- Kernel 4: weighted gather-aggregate of pool_y ---------------- */
__global__ __launch_bounds__(256)
void aggregate_kernel(const float* __restrict__ pool_y,
                      const float* __restrict__ w_ws, const int* __restrict__ ci_ws,
                      float* __restrict__ out)
{
  __shared__ float wl[N_PERIOD * TOPM];
  __shared__ int   il[N_PERIOD * TOPM];
  const int t = threadIdx.x;
  const int b = blockIdx.x;
  if (t < N_PERIOD * TOPM) {
    int p = t / TOPM, m = t % TOPM;
    size_t o = ((size_t)(p * B_SZ + b)) * TOPM + m;
    wl[t] = w_ws[o];
    il[t] = ci_ws[o];
  }
  __syncthreads();
  const int E = PRED_LEN * CHANNELS; /* 768 */
  for (int e = t; e < E; e += 256) {
    float acc = 0.0f;
    for (int k = 0; k < N_PERIOD * TOPM; ++k)
      acc += wl[k] * pool_y[(size_t)il[k] * E + e];
    out[(size_t)b * E + e] = acc * (1.0f / (float)N_PERIOD);
  }
}

extern "C" void kernel_launch(void* const* d_in, const int* in_sizes, int n_in,
                              void* d_out, int out_size, void* d_ws, size_t ws_size,
                              hipStream_t stream)
{
  (void)in_sizes; (void)n_in; (void)out_size; (void)ws_size;
  const int* dataset  = (const int*)d_in[0];
  const int* sensor   = (const int*)d_in[1];
  const int* location = (const int*)d_in[2];
  const int* hour     = (const int*)d_in[3];
  const int* weekday  = (const int*)d_in[4];
  const int* month    = (const int*)d_in[5];
  const int* holiday  = (const int*)d_in[6];
  const int* peak     = (const int*)d_in[7];
  const float* local_state = (const float*)d_in[8];
  const float* pool_ctx    = (const float*)d_in[9];
  const float* pool_y      = (const float*)d_in[10];
  /* params dict flattened in sorted-key order */
  const float* cat_b1       = (const float*)d_in[11];
  const float* cat_b2       = (const float*)d_in[12];
  const float* cat_w1       = (const float*)d_in[13];
  const float* cat_w2       = (const float*)d_in[14];
  const float* emb_dataset  = (const float*)d_in[15];
  const float* emb_holiday  = (const float*)d_in[16];
  const float* emb_hour     = (const float*)d_in[17];
  const float* emb_location = (const float*)d_in[18];
  const float* emb_month    = (const float*)d_in[19];
  const float* emb_peak     = (const float*)d_in[20];
  const float* emb_sensor   = (const float*)d_in[21];
  const float* emb_weekday  = (const float*)d_in[22];
  const float* gate_b1      = (const float*)d_in[23];
  const float* gate_b2      = (const float*)d_in[24];
  const float* gate_w1      = (const float*)d_in[25];
  const float* gate_w2      = (const float*)d_in[26];
  const float* ln_b         = (const float*)d_in[27];
  const float* ln_g         = (const float*)d_in[28];
  const float* loc_b1       = (const float*)d_in[29];
  const float* loc_b2       = (const float*)d_in[30];
  const float* loc_w1       = (const float*)d_in[31];
  const float* loc_w2       = (const float*)d_in[32];

  float* ws = (float*)d_ws;
  const size_t QN = (size_t)N_PERIOD * B_SZ * CTX_DIM;          /* 12288 */
  const size_t SA = (size_t)N_PERIOD * NCHUNK * B_SZ * TOPM;    /* 752640 */
  const size_t WN = (size_t)N_PERIOD * B_SZ * TOPM;             /* 3840 */
  float* q_ws   = ws;
  float* qn_ws  = q_ws + QN;
  float* sa_val = qn_ws + QN;
  int*   sa_idx = (int*)(sa_val + SA);
  float* w_ws   = (float*)(sa_idx + SA);
  int*   ci_ws  = (int*)(w_ws + WN);

  encode_kernel<<<B_SZ, 128, 0, stream>>>(
      dataset, sensor, location, hour, weekday, month, holiday, peak,
      local_state, emb_dataset, emb_sensor, emb_location, emb_hour,
      emb_weekday, emb_month, emb_holiday, emb_peak,
      cat_w1, cat_b1, cat_w2, cat_b2, loc_w1, loc_b1, loc_w2, loc_b2,
      ln_g, ln_b, q_ws, qn_ws);

  dim3 g2(NCHUNK, N_PERIOD);
  sim_topk_kernel<<<g2, 256, 0, stream>>>(pool_ctx, qn_ws, sa_val, sa_idx);

  merge_gate_kernel<<<N_PERIOD * B_SZ, 256, 0, stream>>>(
      sa_val, sa_idx, q_ws, pool_ctx,
      gate_w1, gate_b1, gate_w2, gate_b2, w_ws, ci_ws);

  aggregate_kernel<<<B_SZ, 256, 0, stream>>>(pool_y, w_ws, ci_ws, (float*)d_out);
}